// SSIM_71932112274073
// MI455X (gfx1250) — compile-verified
//
#include <hip/hip_runtime.h>
#include <hip/hip_bf16.h>

// ---- SSIM constants (match reference) ----
#define WIN        7
#define NP_PIX     49.0f
#define COV_NORM   (49.0f / 48.0f)
#define C1_CONST   0.0001f   // (0.01*1.0)^2
#define C2_CONST   0.0009f   // (0.03*1.0)^2

// image geometry
#define IMG_H      256
#define IMG_W      256
#define OUT_H      250       // 256 - 6
#define OUT_W      250
#define PLANE_ELEMS (IMG_H * IMG_W)

// tiling
#define BAND_ROWS  16        // output rows per workgroup
#define CHUNK_COLS 64        // output cols per workgroup
#define N_BANDS    16        // ceil(250/16)
#define N_CHUNKS   4         // ceil(250/64)
#define HALO_ROWS  22        // 16 + 6
#define HALO_COLS  70        // 64 + 6
#define RAW_STRIDE 70        // LDS row stride (floats) for raw tiles (TDM writes contiguous rows)
#define HS_STRIDE  72        // LDS row stride (floats) for hsum arrays (half-wave bank split)
#define HS_ROWS    24        // 22 used, padded to K=24 for WMMA chain

typedef __attribute__((ext_vector_type(2))) float v2f;
typedef __attribute__((ext_vector_type(8))) float v8f;
typedef __attribute__((ext_vector_type(4))) unsigned int u32x4;
typedef __attribute__((ext_vector_type(4))) int i32x4;
typedef __attribute__((ext_vector_type(8))) int i32x8;

// Only probe device builtins in the DEVICE pass — they are not declared for host.
#if defined(__HIP_DEVICE_COMPILE__)
#if !__has_builtin(__builtin_amdgcn_wmma_f32_16x16x4_f32)
#error "missing __builtin_amdgcn_wmma_f32_16x16x4_f32 on gfx1250 device pass"
#endif
#if __has_builtin(__builtin_amdgcn_tensor_load_to_lds) && __has_builtin(__builtin_amdgcn_s_wait_tensorcnt)
#define USE_TDM 1
#endif
#endif

#if defined(USE_TDM)
// Build a 2D-tile Tensor DMA descriptor (D# groups 0 and 1) per CDNA5 ISA §8.3/8.4.
//   group0: [1:0]=count=1, [63:32]=lds_addr, [120:64]=global_addr, [127:126]=type=2
//   group1: [17:16]=data_size=2 (4B), [79:48]=tensor_dim0, [111:80]=tensor_dim1,
//           [127:112]=tile_dim0, [143:128]=tile_dim1, [207:160]=tensor_dim0_stride
__device__ __forceinline__ void tdm_load_2d_f32(
    const float* gptr, unsigned lds_byte_addr,
    unsigned tensor_d0, unsigned tensor_d1,
    unsigned tile_d0, unsigned tile_d1, unsigned stride0)
{
    const unsigned long long ga = (unsigned long long)(uintptr_t)gptr;
    u32x4 g0;
    g0.x = 1u;                                               // count=1, user mode
    g0.y = lds_byte_addr;                                    // lds_addr
    g0.z = (unsigned)(ga & 0xFFFFFFFFu);                     // global_addr[31:0]
    g0.w = (unsigned)((ga >> 32) & 0x01FFFFFFu) | (2u << 30);// global_addr[56:32] | type=2

    i32x8 g1;
    g1[0] = (int)(2u << 16);                                 // wg_mask=0, data_size=4B
    g1[1] = (int)((tensor_d0 & 0xFFFFu) << 16);              // tensor_dim0[15:0] @ bits63:48
    g1[2] = (int)(((tensor_d0 >> 16) & 0xFFFFu) |
                  ((tensor_d1 & 0xFFFFu) << 16));            // td0[31:16] | td1[15:0]
    g1[3] = (int)(((tensor_d1 >> 16) & 0xFFFFu) |
                  ((tile_d0 & 0xFFFFu) << 16));              // td1[31:16] | tile_dim0
    g1[4] = (int)(tile_d1 & 0xFFFFu);                        // tile_dim1 (tile_dim2=0)
    g1[5] = (int)stride0;                                    // tensor_dim0_stride[31:0]
    g1[6] = 0;                                               // stride0[47:32] | stride1[15:0]
    g1[7] = 0;                                               // stride1[47:16]

    const i32x4 gz4 = {0, 0, 0, 0};
#if __clang_major__ >= 23
    const i32x8 gz8 = {0, 0, 0, 0, 0, 0, 0, 0};
    __builtin_amdgcn_tensor_load_to_lds(g0, g1, gz4, gz4, gz8, 0);
#else
    __builtin_amdgcn_tensor_load_to_lds(g0, g1, gz4, gz4, 0);
#endif
}
#endif // USE_TDM

__global__ __launch_bounds__(128) void ssim_tile_kernel(
    const float* __restrict__ pred,
    const float* __restrict__ target,
    float* __restrict__ partials)
{
    __shared__ float ldsP[HALO_ROWS * RAW_STRIDE];
    __shared__ float ldsT[HALO_ROWS * RAW_STRIDE];
    __shared__ float hs[5][HS_ROWS * HS_STRIDE];   // x, y, xx, yy, xy horizontal 7-sums
    __shared__ float red[128];

    const int tid   = threadIdx.x;
    const int wg    = blockIdx.x;
    const int chunk = wg & (N_CHUNKS - 1);
    const int band  = (wg >> 2) & (N_BANDS - 1);
    const int plane = wg >> 6;

    const int oy0 = band * BAND_ROWS;
    const int ox0 = chunk * CHUNK_COLS;
    const float* pP = pred   + (size_t)plane * PLANE_ELEMS;
    const float* pT = target + (size_t)plane * PLANE_ELEMS;

    // ---------- Stage A1: bring the 22x70 halo tile into LDS ----------
#if defined(USE_TDM)
    // Tensor Data Mover: hardware 2D tile walk, tracked on TENSORcnt. One wave
    // issues both loads; rows/cols beyond the tensor extent return zeros, which
    // only feed outputs that are masked out of the reduction below.
    if (tid < 32) {
        const float* gP = pP + (size_t)oy0 * IMG_W + ox0;
        const float* gT = pT + (size_t)oy0 * IMG_W + ox0;
        const unsigned td0 = (unsigned)(IMG_W - ox0);   // remaining width
        const unsigned td1 = (unsigned)(IMG_H - oy0);   // remaining height
        tdm_load_2d_f32(gP, (unsigned)(uintptr_t)ldsP, td0, td1,
                        HALO_COLS, HALO_ROWS, IMG_W);
        tdm_load_2d_f32(gT, (unsigned)(uintptr_t)ldsT, td0, td1,
                        HALO_COLS, HALO_ROWS, IMG_W);
        __builtin_amdgcn_s_wait_tensorcnt(0);
    }
#else
    for (int i = tid; i < HALO_ROWS * HALO_COLS; i += 128) {
        int r  = i / HALO_COLS;
        int c  = i - r * HALO_COLS;
        int gr = oy0 + r; if (gr > IMG_H - 1) gr = IMG_H - 1;
        int gc = ox0 + c; if (gc > IMG_W - 1) gc = IMG_W - 1;
        const size_t off = (size_t)gr * IMG_W + gc;
        ldsP[r * RAW_STRIDE + c] = pP[off];
        ldsT[r * RAW_STRIDE + c] = pT[off];
    }
#endif
    __syncthreads();

    // ---------- Stage A2: horizontal 7-tap sums of the 5 quantities ----------
    for (int i = tid; i < HALO_ROWS * CHUNK_COLS; i += 128) {
        int r = i >> 6;                 // / 64
        int x = i & (CHUNK_COLS - 1);   // % 64
        float sp = 0.f, st = 0.f, spp = 0.f, stt = 0.f, spt = 0.f;
        #pragma unroll
        for (int dx = 0; dx < WIN; ++dx) {
            float p = ldsP[r * RAW_STRIDE + x + dx];
            float t = ldsT[r * RAW_STRIDE + x + dx];
            sp  += p;       st  += t;
            spp += p * p;   stt += t * t;
            spt += p * t;
        }
        hs[0][r * HS_STRIDE + x] = sp;
        hs[1][r * HS_STRIDE + x] = st;
        hs[2][r * HS_STRIDE + x] = spp;
        hs[3][r * HS_STRIDE + x] = stt;
        hs[4][r * HS_STRIDE + x] = spt;
    }
    __syncthreads();

    // ---------- Stage B: vertical 7-tap sums via V_WMMA_F32_16X16X4_F32 ----------
    // Each wave owns one 16x16 output tile. D[m][n] = sum_k band(m,k) * hsum[k][n],
    // band(m,k) = 1 iff m <= k < m+7  (16x24 banded matrix of ones, exact in f32).
    const int lane = tid & 31;
    const int wave = tid >> 5;
    const int txo  = wave * 16;                 // tile column offset within chunk
    const int mrow = lane & 15;                 // A-matrix row (M), also B col (N)
    const int koff = (lane >> 4) << 1;          // 0 for lanes 0-15, 2 for 16-31
    const int colB = txo + (lane & 15);

    v8f acc0 = {}, acc1 = {}, acc2 = {}, acc3 = {}, acc4 = {};

#if defined(__HIP_DEVICE_COMPILE__)
    #pragma unroll
    for (int kb = 0; kb < 6; ++kb) {
        const int k0 = kb * 4 + koff;
        const int k1 = k0 + 1;

        v2f a;
        a.x = (k0 >= mrow && k0 < mrow + WIN) ? 1.0f : 0.0f;
        a.y = (k1 >= mrow && k1 < mrow + WIN) ? 1.0f : 0.0f;

        const bool v0 = (k0 < HALO_ROWS);
        const bool v1 = (k1 < HALO_ROWS);

        v2f b0, b1, b2, b3, b4;
        b0.x = v0 ? hs[0][k0 * HS_STRIDE + colB] : 0.f;
        b0.y = v1 ? hs[0][k1 * HS_STRIDE + colB] : 0.f;
        b1.x = v0 ? hs[1][k0 * HS_STRIDE + colB] : 0.f;
        b1.y = v1 ? hs[1][k1 * HS_STRIDE + colB] : 0.f;
        b2.x = v0 ? hs[2][k0 * HS_STRIDE + colB] : 0.f;
        b2.y = v1 ? hs[2][k1 * HS_STRIDE + colB] : 0.f;
        b3.x = v0 ? hs[3][k0 * HS_STRIDE + colB] : 0.f;
        b3.y = v1 ? hs[3][k1 * HS_STRIDE + colB] : 0.f;
        b4.x = v0 ? hs[4][k0 * HS_STRIDE + colB] : 0.f;
        b4.y = v1 ? hs[4][k1 * HS_STRIDE + colB] : 0.f;

        acc0 = __builtin_amdgcn_wmma_f32_16x16x4_f32(false, a, false, b0, (short)0, acc0, false, false);
        acc1 = __builtin_amdgcn_wmma_f32_16x16x4_f32(false, a, false, b1, (short)0, acc1, false, false);
        acc2 = __builtin_amdgcn_wmma_f32_16x16x4_f32(false, a, false, b2, (short)0, acc2, false, false);
        acc3 = __builtin_amdgcn_wmma_f32_16x16x4_f32(false, a, false, b3, (short)0, acc3, false, false);
        acc4 = __builtin_amdgcn_wmma_f32_16x16x4_f32(false, a, false, b4, (short)0, acc4, false, false);
    }
#endif

    // ---------- SSIM formula per element, mask invalid, reduce ----------
    const float inv_np = 1.0f / NP_PIX;
    float lsum = 0.0f;
    #pragma unroll
    for (int g = 0; g < 8; ++g) {
        const int m   = g + ((lane >> 4) << 3);      // C/D layout: M = g + 8*(lane>=16)
        const int row = oy0 + m;
        const int col = ox0 + txo + (lane & 15);

        const float ux  = acc0[g] * inv_np;
        const float uy  = acc1[g] * inv_np;
        const float uxx = acc2[g] * inv_np;
        const float uyy = acc3[g] * inv_np;
        const float uxy = acc4[g] * inv_np;

        const float vx  = COV_NORM * (uxx - ux * ux);
        const float vy  = COV_NORM * (uyy - uy * uy);
        const float vxy = COV_NORM * (uxy - ux * uy);

        const float A1 = 2.0f * ux * uy + C1_CONST;
        const float A2 = 2.0f * vxy + C2_CONST;
        const float B1 = ux * ux + uy * uy + C1_CONST;
        const float B2 = vx + vy + C2_CONST;
        const float S  = (A1 * A2) / (B1 * B2);

        if (row < OUT_H && col < OUT_W) lsum += S;
    }

    red[tid] = lsum;
    __syncthreads();
    #pragma unroll
    for (int off = 64; off > 0; off >>= 1) {
        if (tid < off) red[tid] += red[tid + off];
        __syncthreads();
    }
    if (tid == 0) partials[wg] = red[0];
}

__global__ __launch_bounds__(256) void ssim_reduce_kernel(
    const float* __restrict__ partials, int n, float inv_count,
    float* __restrict__ out)
{
    __shared__ float red[256];
    float s = 0.0f;
    for (int i = threadIdx.x; i < n; i += 256) s += partials[i];
    red[threadIdx.x] = s;
    __syncthreads();
    #pragma unroll
    for (int off = 128; off > 0; off >>= 1) {
        if (threadIdx.x < off) red[threadIdx.x] += red[threadIdx.x + off];
        __syncthreads();
    }
    if (threadIdx.x == 0) out[0] = red[0] * inv_count;
}

extern "C" void kernel_launch(void* const* d_in, const int* in_sizes, int n_in,
                              void* d_out, int out_size, void* d_ws, size_t ws_size,
                              hipStream_t stream) {
    const float* pred   = (const float*)d_in[0];
    const float* target = (const float*)d_in[1];
    float* out      = (float*)d_out;
    float* partials = (float*)d_ws;

    const int planes = in_sizes[0] / PLANE_ELEMS;           // 384 for the given shape
    const int n_wg   = planes * N_BANDS * N_CHUNKS;         // 24576
    const float inv_count = 1.0f / ((float)planes * (float)OUT_H * (float)OUT_W);

    ssim_tile_kernel<<<n_wg, 128, 0, stream>>>(pred, target, partials);
    ssim_reduce_kernel<<<1, 256, 0, stream>>>(partials, n_wg, inv_count, out);
}